// TotalLoss_58437325029969
// MI455X (gfx1250) — compile-verified
//
#include <hip/hip_runtime.h>
#include <hip/hip_bf16.h>
#include <climits>

#define CEPS 1.1920928955078125e-07f   // np.finfo(np.float32).eps = 2^-23

typedef float v2f __attribute__((ext_vector_type(2)));
typedef float v8f __attribute__((ext_vector_type(8)));

// ---------------- workspace layout (in floats) ----------------
// NUM:    [0, 696320)            (b,scale,pol,dir)-ordered grids
// DEN:    [696320, 1392640)      same internal ordering as NUM
// PARAMS: [1392640, 1392768)     32 x float4 {anchor_t, s, +/-EPS, 0}
// IDX:    8 ints after params    {firstPos,lastPos,firstNeg,lastNeg} x B
#define NUM_CELLS   696320        // = 2 * 4 * 87040... per-b stride below
#define B_STRIDE    348160        // 4 * (1024+4096+16384+65536)
#define ZERO_FLOATS 1392640       // NUM + DEN

__device__ __constant__ int c_prefWH4[4] = {0, 4096, 20480, 86016}; // 4*prefix(WH)

// ---------------- init: zero accumulators, out, sentinels ----------------
__global__ __launch_bounds__(256) void init_kernel(float* ws, int* idx, float* out) {
    int i = blockIdx.x * blockDim.x + threadIdx.x;
    int stride = gridDim.x * blockDim.x;
    for (int j = i; j < ZERO_FLOATS; j += stride) ws[j] = 0.0f;
    if (i == 0) out[0] = 0.0f;
    if (i < 8) idx[i] = (i & 1) ? -1 : INT_MAX;  // even slots: min-sentinel, odd: max-sentinel
}

// ---------------- first/last index of each polarity ----------------
__global__ __launch_bounds__(256) void minmax_kernel(const int* __restrict__ ps, int N, int* idx) {
    int b = blockIdx.y;
    const int* p = ps + (long)b * N;
    int i0 = blockIdx.x * blockDim.x + threadIdx.x;
    int stride = gridDim.x * blockDim.x;
    int fp = INT_MAX, lp = -1, fn = INT_MAX, ln = -1;
    for (int i = i0; i < N; i += stride) {
        int v = p[i];
        if (v == 1)       { fp = min(fp, i); lp = max(lp, i); }
        else if (v == -1) { fn = min(fn, i); ln = max(ln, i); }
    }
    for (int o = 16; o; o >>= 1) {
        fp = min(fp, __shfl_xor(fp, o, 32));
        lp = max(lp, __shfl_xor(lp, o, 32));
        fn = min(fn, __shfl_xor(fn, o, 32));
        ln = max(ln, __shfl_xor(ln, o, 32));
    }
    if ((threadIdx.x & 31) == 0) {
        atomicMin(&idx[b * 4 + 0], fp);
        atomicMax(&idx[b * 4 + 1], lp);
        atomicMin(&idx[b * 4 + 2], fn);
        atomicMax(&idx[b * 4 + 3], ln);
    }
}

// ---------------- collapse t-renormalization chain to scalars ----------------
__global__ void params_kernel(const float* __restrict__ ts, int N,
                              const int* __restrict__ idx, float4* params) {
    int b = threadIdx.x;
    if (b >= 2) return;
    const float* t = ts + (long)b * N;
    float t0 = t[0], tN = t[N - 1];
    int fp = idx[b*4+0]; if (fp == INT_MAX) fp = 0;
    int lp = idx[b*4+1]; if (lp < 0)        lp = N - 1;
    int fn = idx[b*4+2]; if (fn == INT_MAX) fn = 0;
    int ln = idx[b*4+3]; if (ln < 0)        ln = N - 1;
    float tfp = t[fp], tlp = t[lp], tfn = t[fn], tln = t[ln];
    float d = tN - t0;
    float s = 1.0f / (d + CEPS);               // normalization slope at scale 0
    for (int sc = 0; sc < 4; ++sc) {
        if (sc > 0) { float r = d * s; s = s / (r + CEPS); }
        // index: ((b*4+sc)*2 + pol)*2 + dir ; pol0=pos, pol1=neg; dir0=fwd, dir1=bwd
        params[((b*4+sc)*2+0)*2+0] = make_float4(tlp, s,  CEPS, 0.0f);
        params[((b*4+sc)*2+0)*2+1] = make_float4(tfp, s, -CEPS, 0.0f);
        params[((b*4+sc)*2+1)*2+0] = make_float4(tln, s,  CEPS, 0.0f);
        params[((b*4+sc)*2+1)*2+1] = make_float4(tfn, s, -CEPS, 0.0f);
    }
}

// ---------------- event scatter-accumulate (all scales, both dirs) ----------------
__global__ __launch_bounds__(256) void accum_kernel(
    const float* __restrict__ flow0, const float* __restrict__ flow1,
    const float* __restrict__ flow2, const float* __restrict__ flow3,
    const int* __restrict__ xs, const int* __restrict__ ys,
    const float* __restrict__ ts, const int* __restrict__ ps, int N,
    float* __restrict__ NUM, float* __restrict__ DEN,
    const float4* __restrict__ params)
{
    __shared__ float lds[8192];                 // scale-0: 4 combos x {den,num} x 1024
    int b = blockIdx.y;
    for (int j = threadIdx.x; j < 8192; j += 256) lds[j] = 0.0f;
    __syncthreads();

    const float4* P = params + b * 16;
    int i0 = blockIdx.x * blockDim.x + threadIdx.x;
    int stride = gridDim.x * blockDim.x;
    long eb = (long)b * N;

    for (int i = i0; i < N; i += stride) {
        int   x = xs[eb + i];
        int   y = ys[eb + i];
        float t = ts[eb + i];
        int   p = ps[eb + i];
        int pol = (p == 1) ? 0 : 1;

        #pragma unroll
        for (int sc = 0; sc < 4; ++sc) {
            const int shift = 3 - sc;
            const int W = 32 << sc;
            const int HW = W * W;
            int xi = x >> shift, yi = y >> shift;   // == int(float(x)/2^shift), x>=0
            const float* fl = (sc == 0) ? flow0 : (sc == 1) ? flow1 : (sc == 2) ? flow2 : flow3;
            float fx = fl[((long)(b * 2 + 0)) * HW + yi * W + xi];
            float fy = fl[((long)(b * 2 + 1)) * HW + yi * W + xi];

            #pragma unroll
            for (int dir = 0; dir < 2; ++dir) {
                float4 pr = P[(sc * 2 + pol) * 2 + dir];
                float t_ = (pr.x - t) * pr.y + pr.z;
                float xf = fminf(fmaxf((float)xi + t_ * fx, 0.0f), (float)(W - 1));
                float yf = fminf(fmaxf((float)yi + t_ * fy, 0.0f), (float)(W - 1));
                float x0 = floorf(xf), y0 = floorf(yf);
                float x0_ = xf - x0,  y0_ = yf - y0;
                float x1_ = 1.0f - x0_, y1_ = 1.0f - y0_;
                float Ra = x0_ * y0_ + CEPS, Rb = x1_ * y0_ + CEPS;
                float Rc = x0_ * y1_ + CEPS, Rd = x1_ * y1_ + CEPS;
                int x0i = (int)x0, y0i = (int)y0;
                int x1i = min(x0i + 1, W - 1), y1i = min(y0i + 1, W - 1);
                int Ia = x1i + y1i * W, Ib = x0i + y1i * W;
                int Ic = x1i + y0i * W, Id = x0i + y0i * W;
                if (sc == 0) {
                    float* dl = &lds[((pol * 2 + dir) * 2) * 1024];
                    float* nl = dl + 1024;
                    atomicAdd(dl + Ia, 1.0f);      atomicAdd(dl + Ib, 1.0f);
                    atomicAdd(dl + Ic, 1.0f);      atomicAdd(dl + Id, 1.0f);
                    atomicAdd(nl + Ia, Ra * t_);   atomicAdd(nl + Ib, Rb * t_);
                    atomicAdd(nl + Ic, Rc * t_);   atomicAdd(nl + Id, Rd * t_);
                } else {
                    int base = b * B_STRIDE + c_prefWH4[sc] + (pol * 2 + dir) * HW;
                    float* dg = DEN + base;
                    float* ng = NUM + base;
                    atomicAdd(dg + Ia, 1.0f);      atomicAdd(dg + Ib, 1.0f);
                    atomicAdd(dg + Ic, 1.0f);      atomicAdd(dg + Id, 1.0f);
                    atomicAdd(ng + Ia, Ra * t_);   atomicAdd(ng + Ib, Rb * t_);
                    atomicAdd(ng + Ic, Rc * t_);   atomicAdd(ng + Id, Rd * t_);
                }
            }
        }
    }
    __syncthreads();
    // flush scale-0 LDS tiles (one global atomic per nonzero cell per block)
    for (int j = threadIdx.x; j < 8192; j += 256) {
        float v = lds[j];
        if (v != 0.0f) {
            int plane = j >> 10, cell = j & 1023;
            int q = plane >> 1, isNum = plane & 1;
            int base = b * B_STRIDE + q * 1024 + cell;
            atomicAdd((isNum ? NUM : DEN) + base, v);
        }
    }
}

// ---------------- dense reduction via v_wmma_f32_16x16x4_f32 ----------------
// total (=696320) is an exact multiple of 64; every WMMA issues with full EXEC.
__global__ __launch_bounds__(256) void reduce_wmma_kernel(
    const float* __restrict__ NUM, const float* __restrict__ DEN,
    float* __restrict__ out, int total)
{
    int lane   = threadIdx.x & 31;
    int wave   = (blockIdx.x * blockDim.x + threadIdx.x) >> 5;
    int nwaves = (gridDim.x * blockDim.x) >> 5;
    v8f c = {0.0f, 0.0f, 0.0f, 0.0f, 0.0f, 0.0f, 0.0f, 0.0f};
    v2f ones; ones.x = 1.0f; ones.y = 1.0f;
    for (int base = wave * 64; base < total; base += nwaves * 64) {   // wave-uniform
        int j = base + lane * 2;
        float n0 = NUM[j],     d0 = DEN[j];
        float n1 = NUM[j + 1], d1 = DEN[j + 1];
        float v0 = n0 / (d0 + CEPS);
        float v1 = n1 / (d1 + CEPS);
        v2f a; a.x = v0 * v0; a.y = v1 * v1;
        // D = A x ones + C : exact f32 row-sums accumulated into C
        c = __builtin_amdgcn_wmma_f32_16x16x4_f32(false, a, false, ones,
                                                  (short)0, c, false, false);
    }
    float psum = c[0] + c[1] + c[2] + c[3] + c[4] + c[5] + c[6] + c[7];
    psum += __shfl_xor(psum, 16, 32);          // lanes l and l+16 complete a column
    if (lane == 0) atomicAdd(out, psum);
}

// ---------------- smoothness (charbonnier over 8-connected diffs) ----------------
__global__ __launch_bounds__(256) void smooth_kernel(
    const float* __restrict__ f0, const float* __restrict__ f1,
    const float* __restrict__ f2, const float* __restrict__ f3,
    float* __restrict__ out)
{
    int lvl = blockIdx.y;
    const float* f = (lvl == 0) ? f0 : (lvl == 1) ? f1 : (lvl == 2) ? f2 : f3;
    int H = 32 << lvl;
    int HW = H * H;
    int total = 4 * HW;                         // B(2) * channels(2) * H * W
    float inv12 = 1.0f / (float)(4 * (H - 1) * H);        // vert & horiz counts
    float inv34 = 1.0f / (float)(4 * (H - 1) * (H - 1));  // diagonal counts
    int i0 = blockIdx.x * blockDim.x + threadIdx.x;
    int stride = gridDim.x * blockDim.x;
    float acc = 0.0f;
    for (int i = i0; i < total; i += stride) {
        int bc = i / HW, r = i - bc * HW;
        int h = r / H,  w = r - h * H;
        const float* p = f + (long)bc * HW;
        float v = p[r];
        if (h < H - 1) { float d = p[r + H] - v;     acc += powf(d * d + 1e-6f, 0.45f) * inv12; }
        if (w < H - 1) { float d = p[r + 1] - v;     acc += powf(d * d + 1e-6f, 0.45f) * inv12; }
        if (h < H - 1 && w < H - 1) {
            float d1 = p[r + H + 1] - v;             acc += powf(d1 * d1 + 1e-6f, 0.45f) * inv34;
            float d2 = p[r + 1] - p[r + H];          acc += powf(d2 * d2 + 1e-6f, 0.45f) * inv34;
        }
    }
    acc *= 0.5f;                                // SMOOTHNESS_WEIGHT
    for (int o = 16; o; o >>= 1) acc += __shfl_down(acc, o, 32);
    if ((threadIdx.x & 31) == 0) atomicAdd(out, acc);
}

// ---------------- launcher ----------------
extern "C" void kernel_launch(void* const* d_in, const int* in_sizes, int n_in,
                              void* d_out, int out_size, void* d_ws, size_t ws_size,
                              hipStream_t stream) {
    const float* flow0 = (const float*)d_in[0];
    const float* flow1 = (const float*)d_in[1];
    const float* flow2 = (const float*)d_in[2];
    const float* flow3 = (const float*)d_in[3];
    const int*   xs    = (const int*)d_in[4];
    const int*   ys    = (const int*)d_in[5];
    const float* ts    = (const float*)d_in[6];
    const int*   ps    = (const int*)d_in[7];
    const int N = in_sizes[4] / 2;              // B = 2

    float*  ws     = (float*)d_ws;
    float*  NUM    = ws;
    float*  DEN    = ws + NUM_CELLS;
    float4* params = (float4*)(ws + ZERO_FLOATS);
    int*    idx    = (int*)(ws + ZERO_FLOATS + 128);
    float*  out    = (float*)d_out;

    init_kernel<<<2048, 256, 0, stream>>>(ws, idx, out);
    minmax_kernel<<<dim3(512, 2), 256, 0, stream>>>(ps, N, idx);
    params_kernel<<<1, 32, 0, stream>>>(ts, N, idx, params);
    accum_kernel<<<dim3(256, 2), 256, 0, stream>>>(flow0, flow1, flow2, flow3,
                                                   xs, ys, ts, ps, N, NUM, DEN, params);
    reduce_wmma_kernel<<<128, 256, 0, stream>>>(NUM, DEN, out, NUM_CELLS);
    smooth_kernel<<<dim3(64, 4), 256, 0, stream>>>(flow0, flow1, flow2, flow3, out);
}